// DecorelationNormalization_11965778886865
// MI455X (gfx1250) — compile-verified
//
#include <hip/hip_runtime.h>

typedef float v2f __attribute__((ext_vector_type(2)));
typedef float v8f __attribute__((ext_vector_type(8)));

#define C_CH   256
#define MG     16
#define N_POS  200704            // 64*56*56, divisible by 16
#define N_TILES (N_POS / 16)
#define EPS_F  1e-3f
#define PITCH  260               // 260 % 64 = 4 -> conflict-free strided reads; rows 16B-aligned (1040B)

// ---------------------------------------------------------------------------
// CDNA5 async global->LDS copy (ASYNCcnt-tracked), per-lane 16B.
// VDST = LDS byte offset (workgroup-relative), VADDR = 64-bit global address.
// ---------------------------------------------------------------------------
__device__ __forceinline__ void async_copy_b128(const float* gsrc, const float* lds_dst) {
    unsigned lds_off = (unsigned)(unsigned long long)lds_dst;  // low 32 bits = LDS offset
    asm volatile("global_load_async_to_lds_b128 %0, %1, off"
                 :: "v"(lds_off), "v"(gsrc)
                 : "memory");
}
__device__ __forceinline__ void wait_async_le(int) {}  // (doc helper; real waits below)
__device__ __forceinline__ void s_wait_asynccnt_2() {
    asm volatile("s_wait_asynccnt 0x2" ::: "memory");
}
__device__ __forceinline__ void s_wait_asynccnt_0() {
    asm volatile("s_wait_asynccnt 0x0" ::: "memory");
}

// ---------------------------------------------------------------------------
// Pass A: fused per-channel sums + per-group raw second moment E[x x^T]*N
// One wave per group (16 waves / 512-thread block). A-layout load is a 64B
// contiguous segment per half-wave -> coalesced. A == B^T in registers.
// ---------------------------------------------------------------------------
__global__ void dn_stats(const float* __restrict__ x,
                         float* __restrict__ sums,   // [256]
                         float* __restrict__ xxT) {  // [16][16][16]
    const int t    = threadIdx.x;
    const int g    = t >> 5;          // wave id == group id
    const int lane = t & 31;
    const int half = lane >> 4;
    const int m    = lane & 15;
    const int ch   = g * MG + m;

    v8f acc0 = {}; v8f acc1 = {};
    float ssum = 0.f;

    for (int tile = blockIdx.x; tile < N_TILES; tile += gridDim.x) {
        const int base = tile * 16;
#pragma unroll
        for (int kk = 0; kk < 4; ++kk) {
            const int p = base + kk * 4 + half * 2;   // K = kk*4 + half*2 + {0,1}
            v2f a;
            a.x = x[p * C_CH + ch];
            a.y = x[(p + 1) * C_CH + ch];
            ssum += a.x + a.y;
            if (kk & 1)
                acc1 = __builtin_amdgcn_wmma_f32_16x16x4_f32(false, a, false, a,
                                                             (short)0, acc1, false, false);
            else
                acc0 = __builtin_amdgcn_wmma_f32_16x16x4_f32(false, a, false, a,
                                                             (short)0, acc0, false, false);
        }
    }
    v8f acc = acc0 + acc1;
#pragma unroll
    for (int i = 0; i < 8; ++i)      // D layout: row = i + 8*half, col = lane&15
        atomicAdd(&xxT[g * 256 + (i + half * 8) * 16 + m], acc[i]);
    atomicAdd(&sums[ch], ssum);
}

// ---------------------------------------------------------------------------
// Pass B: sigma = (1-eps)*(E[xx^T]-mm^T) + eps*I ; 5 Newton-Schulz iterations
// 16 blocks (one per group) x 256 threads (one per matrix element).
// ---------------------------------------------------------------------------
__global__ void dn_newton(const float* __restrict__ sums,
                          const float* __restrict__ xxT,
                          float* __restrict__ means,   // [256]
                          float* __restrict__ wmat) {  // [16][16][16]
    __shared__ float S[256], P[256], T1[256], T2[256], mean_sh[16];
    __shared__ float tr_sh;
    const int g = blockIdx.x;
    const int t = threadIdx.x;
    const int i = t >> 4, j = t & 15;
    const float invN = 1.0f / (float)N_POS;

    if (t < 16) mean_sh[t] = sums[g * 16 + t] * invN;
    __syncthreads();

    float raw = xxT[g * 256 + t] * invN - mean_sh[i] * mean_sh[j];
    S[t] = (1.0f - EPS_F) * raw + (i == j ? EPS_F : 0.0f);
    __syncthreads();

    if (t == 0) {
        float tr = 0.f;
        for (int k = 0; k < 16; ++k) tr += S[k * 16 + k];
        tr_sh = tr;
    }
    __syncthreads();

    S[t] = S[t] / tr_sh;             // sigma_n (own element only)
    P[t] = (i == j) ? 1.0f : 0.0f;
    __syncthreads();

    for (int it = 0; it < 5; ++it) {
        float v = 0.f;
        for (int k = 0; k < 16; ++k) v += P[i * 16 + k] * P[k * 16 + j];
        T1[t] = v;  __syncthreads();                 // P^2
        v = 0.f;
        for (int k = 0; k < 16; ++k) v += T1[i * 16 + k] * P[k * 16 + j];
        T2[t] = v;  __syncthreads();                 // P^3
        v = 0.f;
        for (int k = 0; k < 16; ++k) v += T2[i * 16 + k] * S[k * 16 + j];
        float pnew = 1.5f * P[t] - 0.5f * v;         // 1.5P - 0.5 P^3 sigma_n
        __syncthreads();
        P[t] = pnew;
        __syncthreads();
    }
    wmat[g * 256 + t] = P[t] * rsqrtf(tr_sh);
    if (t < 16) means[g * 16 + t] = mean_sh[t];
}

// ---------------------------------------------------------------------------
// Pass C: out = W @ (x - mean). Double-buffered async global->LDS tile
// prefetch (ASYNCcnt) overlaps the next tile's DMA with this tile's WMMAs.
// Wave g computes its group's 16x16 output tile with 4 f32 WMMAs.
// ---------------------------------------------------------------------------
__global__ void dn_whiten(const float* __restrict__ x,
                          const float* __restrict__ means,
                          const float* __restrict__ wmat,
                          float* __restrict__ out) {
    __shared__ float Ft[2][16 * PITCH];   // double-buffered raw-x tiles
    __shared__ float Ot[16 * PITCH];      // output staging tile

    const int t    = threadIdx.x;     // 512 threads = 16 waves
    const int g    = t >> 5;
    const int lane = t & 31;
    const int half = lane >> 4;
    const int m    = lane & 15;       // A: row of W ; B/D: position within tile

    // per-thread slice of the cooperative 16KB tile copy (2 x b128 per thread)
    const int f4a = t,        pa = f4a >> 6, ca = (f4a & 63) * 4;
    const int f4b = t + 512,  pb = f4b >> 6, cb = (f4b & 63) * 4;

    // Preload W in A layout and the matching mean pair per K-chunk.
    v2f a[4], mn[4];
#pragma unroll
    for (int kk = 0; kk < 4; ++kk) {
        const int kb = kk * 4 + half * 2;
        a[kk].x  = wmat[g * 256 + m * 16 + kb];
        a[kk].y  = wmat[g * 256 + m * 16 + kb + 1];
        mn[kk].x = means[g * MG + kb];
        mn[kk].y = means[g * MG + kb + 1];
    }

    int tile = blockIdx.x;
    int cur  = 0;
    if (tile < N_TILES) {   // uniform across the block
        const int base = tile * 16;
        async_copy_b128(x + (base + pa) * C_CH + ca, &Ft[0][pa * PITCH + ca]);
        async_copy_b128(x + (base + pb) * C_CH + cb, &Ft[0][pb * PITCH + cb]);
    }

    for (; tile < N_TILES; tile += gridDim.x) {
        const int base = tile * 16;
        const int next = tile + gridDim.x;
        if (next < N_TILES) {         // prefetch next tile into alternate buffer
            const int nbase = next * 16;
            async_copy_b128(x + (nbase + pa) * C_CH + ca, &Ft[cur ^ 1][pa * PITCH + ca]);
            async_copy_b128(x + (nbase + pb) * C_CH + cb, &Ft[cur ^ 1][pb * PITCH + cb]);
            s_wait_asynccnt_2();      // current tile's 2 copies retired (in-order)
        } else {
            s_wait_asynccnt_0();
        }
        __syncthreads();              // all waves' copies of current tile visible

        v8f acc0 = {}; v8f acc1 = {};
#pragma unroll
        for (int kk = 0; kk < 4; ++kk) {
            const int kb = kk * 4 + half * 2;
            v2f b;                    // B: K = channel, N = position m; center here
            b.x = Ft[cur][m * PITCH + g * MG + kb]     - mn[kk].x;
            b.y = Ft[cur][m * PITCH + g * MG + kb + 1] - mn[kk].y;
            if (kk & 1)
                acc1 = __builtin_amdgcn_wmma_f32_16x16x4_f32(false, a[kk], false, b,
                                                             (short)0, acc1, false, false);
            else
                acc0 = __builtin_amdgcn_wmma_f32_16x16x4_f32(false, a[kk], false, b,
                                                             (short)0, acc0, false, false);
        }
        v8f acc = acc0 + acc1;
#pragma unroll
        for (int i = 0; i < 8; ++i)   // D: M = out-channel, N = position
            Ot[m * PITCH + g * MG + i + half * 8] = acc[i];
        __syncthreads();

        // coalesced float4 store of the staged output tile
        {
            const float* s0 = &Ot[pa * PITCH + ca];
            float4 v0; v0.x = s0[0]; v0.y = s0[1]; v0.z = s0[2]; v0.w = s0[3];
            *(float4*)(out + (base + pa) * C_CH + ca) = v0;
            const float* s1 = &Ot[pb * PITCH + cb];
            float4 v1; v1.x = s1[0]; v1.y = s1[1]; v1.z = s1[2]; v1.w = s1[3];
            *(float4*)(out + (base + pb) * C_CH + cb) = v1;
        }
        __syncthreads();              // Ot reusable; Ft[cur] reusable next round
        cur ^= 1;
    }
}

// ---------------------------------------------------------------------------
extern "C" void kernel_launch(void* const* d_in, const int* in_sizes, int n_in,
                              void* d_out, int out_size, void* d_ws, size_t ws_size,
                              hipStream_t stream) {
    const float* x = (const float*)d_in[0];
    float* out = (float*)d_out;
    float* ws  = (float*)d_ws;

    float* sums  = ws;                       // 256
    float* xxT   = ws + 256;                 // 4096
    float* means = ws + 256 + 4096;          // 256
    float* wmat  = ws + 256 + 4096 + 256;    // 4096

    hipMemsetAsync(ws, 0, (256 + 4096) * sizeof(float), stream);
    dn_stats <<<512,  512, 0, stream>>>(x, sums, xxT);
    dn_newton<<<16,   256, 0, stream>>>(sums, xxT, means, wmat);
    dn_whiten<<<2048, 512, 0, stream>>>(x, means, wmat, out);
}